// GCNBlock_45887430590920
// MI455X (gfx1250) — compile-verified
//
#include <hip/hip_runtime.h>

typedef __attribute__((ext_vector_type(2))) float v2f;
typedef __attribute__((ext_vector_type(8))) float v8f;

#define N_NODES 100000
#define HID 64
#define IN_CH 128
#define N_EDGES 1000000

// ---------------------------------------------------------------------------
// Degree / normalization: deg = 1 (self loop) + in-degree(dst); dinv = rsqrt
// ---------------------------------------------------------------------------
__global__ void deg_init_kernel(float* __restrict__ deg) {
  int i = blockIdx.x * 256 + threadIdx.x;
  if (i < N_NODES) deg[i] = 1.0f;
}

__global__ void deg_count_kernel(const long long* __restrict__ dst,
                                 float* __restrict__ deg) {
  int e = blockIdx.x * 256 + threadIdx.x;
  if (e < N_EDGES) atomicAdd(&deg[(int)dst[e]], 1.0f);
}

__global__ void deg_rsqrt_kernel(float* __restrict__ deg) {
  int i = blockIdx.x * 256 + threadIdx.x;
  if (i < N_NODES) {
    float d = deg[i];
    deg[i] = (d > 0.0f) ? rsqrtf(d) : 0.0f;
  }
}

// ---------------------------------------------------------------------------
// GEMM via V_WMMA_F32_16X16X4_F32 (full fp32 matrix pipe).
//   C[M x 64] = A[M x K] * W[K x 64] (+ bias)      M % 16 == 0
// Block = 128 threads (4 waves). Wave w owns the 16x16 tile at columns w*16.
// A fragment (16x4 f32):  lane<16 -> K = k+0,k+1 ; lane>=16 -> K = k+2,k+3
// B fragment (4x16 f32):  V0 = rows k+2*hi, V1 = rows k+1+2*hi, N = lane&15
// C/D (16x16 f32, 8 VGPRs): vgpr v, lane<16 -> M=v, lane>=16 -> M=v+8.
// ---------------------------------------------------------------------------
template <int K>
__global__ void gemm_wmma_kernel(const float* __restrict__ A,
                                 const float* __restrict__ W,
                                 const float* __restrict__ bias,
                                 float* __restrict__ C,
                                 float* __restrict__ C2) {
  const int wave = threadIdx.x >> 5;   // 0..3 -> N tile
  const int lane = threadIdx.x & 31;
  const int l16  = lane & 15;
  const int hi   = lane >> 4;          // 0: lanes 0-15, 1: lanes 16-31
  const int row0 = blockIdx.x * 16;
  const int col0 = wave * 16;

  v8f acc = {};
#pragma unroll
  for (int k = 0; k < K; k += 4) {
    // A: row = row0 + l16, two consecutive K values per lane
    const float* ap = A + (size_t)(row0 + l16) * K + (k + 2 * hi);
    v2f a;
    a.x = ap[0];
    a.y = ap[1];
    // B: col = col0 + l16, rows k+2*hi and k+2*hi+1 (row stride = HID)
    const float* bp = W + (size_t)(k + 2 * hi) * HID + (col0 + l16);
    v2f b;
    b.x = bp[0];
    b.y = bp[HID];
    acc = __builtin_amdgcn_wmma_f32_16x16x4_f32(
        /*neg_a=*/false, a, /*neg_b=*/false, b,
        /*c_mod=*/(short)0, acc, /*reuse_a=*/false, /*reuse_b=*/false);
  }

  const float bcol = bias ? bias[col0 + l16] : 0.0f;
#pragma unroll
  for (int v = 0; v < 8; ++v) {
    const int m = v + 8 * hi;
    const size_t idx = (size_t)(row0 + m) * HID + (col0 + l16);
    const float val = acc[v] + bcol;
    C[idx] = val;
    if (C2) C2[idx] = val;
  }
}

// ---------------------------------------------------------------------------
// agg[i][c] = hw[i][c] * dinv[i]^2   (self-loop term; also initializes agg)
// ---------------------------------------------------------------------------
__global__ void self_loop_kernel(const float* __restrict__ hw,
                                 const float* __restrict__ dinv,
                                 float* __restrict__ agg) {
  int idx = blockIdx.x * 256 + threadIdx.x;   // over N_NODES*HID
  int r = idx >> 6;
  if (r < N_NODES) {
    float di = dinv[r];
    agg[idx] = hw[idx] * (di * di);
  }
}

// ---------------------------------------------------------------------------
// Edge scatter: 16 lanes per edge, float4 per lane (coalesced row access).
// agg[dst] += hw[src] * dinv[src]*dinv[dst]
// ---------------------------------------------------------------------------
__global__ void scatter_edges_kernel(const long long* __restrict__ src,
                                     const long long* __restrict__ dst,
                                     const float* __restrict__ hw,
                                     const float* __restrict__ dinv,
                                     float* agg) {
  long long gid = (long long)blockIdx.x * 256 + threadIdx.x;
  int e = (int)(gid >> 4);
  if (e >= N_EDGES) return;
  int c0 = ((int)gid & 15) << 2;
  int s = (int)src[e];
  int d = (int)dst[e];
  float norm = dinv[s] * dinv[d];
  const float4 hv = *(const float4*)(hw + (size_t)s * HID + c0);
  float* ap = agg + (size_t)d * HID + c0;
  atomicAdd(ap + 0, hv.x * norm);
  atomicAdd(ap + 1, hv.y * norm);
  atomicAdd(ap + 2, hv.z * norm);
  atomicAdd(ap + 3, hv.w * norm);
}

// ---------------------------------------------------------------------------
// Fused bias + LayerNorm + (ReLU + residual). Warp (wave32) per row,
// 2 channels per lane, __shfl_xor tree reduction.
// fuse_relu_res: 1 -> y = relu(LN(x+b)*g+beta) + res; write out_h and res_out
//                0 -> y = LN(x+b)*g+beta; write out_h only
// NOTE: agg may alias out_h (in-place safe: each thread reads before writing
// its own elements only).
// ---------------------------------------------------------------------------
__global__ void ln_kernel(const float* agg, const float* __restrict__ bias,
                          const float* __restrict__ g,
                          const float* __restrict__ beta,
                          const float* res_in, float* out_h, float* res_out,
                          int fuse_relu_res) {
  int warp = threadIdx.x >> 5;
  int lane = threadIdx.x & 31;
  int row = blockIdx.x * 8 + warp;
  if (row >= N_NODES) return;
  int c = lane * 2;

  float2 v = *(const float2*)(agg + (size_t)row * HID + c);
  v.x += bias[c];
  v.y += bias[c + 1];

  float s  = v.x + v.y;
  float sq = v.x * v.x + v.y * v.y;
#pragma unroll
  for (int off = 16; off >= 1; off >>= 1) {
    s  += __shfl_xor(s, off, 32);
    sq += __shfl_xor(sq, off, 32);
  }
  float mu  = s * (1.0f / 64.0f);
  float var = sq * (1.0f / 64.0f) - mu * mu;
  float inv = rsqrtf(var + 1e-5f);

  float y0 = (v.x - mu) * inv * g[c] + beta[c];
  float y1 = (v.y - mu) * inv * g[c + 1] + beta[c + 1];

  if (fuse_relu_res) {
    float2 r = *(const float2*)(res_in + (size_t)row * HID + c);
    y0 = fmaxf(y0, 0.0f) + r.x;
    y1 = fmaxf(y1, 0.0f) + r.y;
    float2 o;
    o.x = y0;
    o.y = y1;
    *(float2*)(res_out + (size_t)row * HID + c) = o;
  }
  float2 o;
  o.x = y0;
  o.y = y1;
  *(float2*)(out_h + (size_t)row * HID + c) = o;
}

// ---------------------------------------------------------------------------
extern "C" void kernel_launch(void* const* d_in, const int* in_sizes, int n_in,
                              void* d_out, int out_size, void* d_ws,
                              size_t ws_size, hipStream_t stream) {
  const float* x           = (const float*)d_in[0];
  const long long* edge    = (const long long*)d_in[1];
  const long long* src     = edge;               // edge_index[0]
  const long long* dst     = edge + N_EDGES;     // edge_index[1]
  const float* proj_w      = (const float*)d_in[2];
  const float* proj_b      = (const float*)d_in[3];
  const float* W[3]  = {(const float*)d_in[4], (const float*)d_in[8],
                        (const float*)d_in[12]};
  const float* B[3]  = {(const float*)d_in[5], (const float*)d_in[9],
                        (const float*)d_in[13]};
  const float* G[3]  = {(const float*)d_in[6], (const float*)d_in[10],
                        (const float*)d_in[14]};
  const float* BE[3] = {(const float*)d_in[7], (const float*)d_in[11],
                        (const float*)d_in[15]};

  // Workspace carve-out
  char* ws = (char*)d_ws;
  float* dinv = (float*)ws;
  size_t off = ((size_t)N_NODES * sizeof(float) + 255) & ~(size_t)255;
  float* h   = (float*)(ws + off); off += (size_t)N_NODES * HID * sizeof(float);
  float* res = (float*)(ws + off); off += (size_t)N_NODES * HID * sizeof(float);
  float* hw  = (float*)(ws + off);
  float* agg = h;  // h is dead after each layer's GEMM -> reuse as agg target

  const int TB = 256;
  // 1) symmetric degree normalization (graph is layer-invariant)
  deg_init_kernel<<<(N_NODES + TB - 1) / TB, TB, 0, stream>>>(dinv);
  deg_count_kernel<<<(N_EDGES + TB - 1) / TB, TB, 0, stream>>>(dst, dinv);
  deg_rsqrt_kernel<<<(N_NODES + TB - 1) / TB, TB, 0, stream>>>(dinv);

  // 2) input projection (WMMA f32), residual = h
  gemm_wmma_kernel<IN_CH><<<N_NODES / 16, 128, 0, stream>>>(x, proj_w, proj_b,
                                                            h, res);

  // 3) three GCN layers
  for (int l = 0; l < 3; ++l) {
    gemm_wmma_kernel<HID><<<N_NODES / 16, 128, 0, stream>>>(h, W[l], nullptr,
                                                            hw, nullptr);
    self_loop_kernel<<<(N_NODES * HID) / TB, TB, 0, stream>>>(hw, dinv, agg);
    scatter_edges_kernel<<<(N_EDGES * 16) / TB, TB, 0, stream>>>(src, dst, hw,
                                                                 dinv, agg);
    if (l < 2) {
      ln_kernel<<<N_NODES / 8, TB, 0, stream>>>(agg, B[l], G[l], BE[l], res, h,
                                                res, 1);
    } else {
      ln_kernel<<<N_NODES / 8, TB, 0, stream>>>(agg, B[l], G[l], BE[l], res,
                                                (float*)d_out, res, 0);
    }
  }
}